// GNNModel_14482629722973
// MI455X (gfx1250) — compile-verified
//
#include <hip/hip_runtime.h>
#include <hip/hip_bf16.h>

typedef float v2f __attribute__((ext_vector_type(2)));
typedef float v8f __attribute__((ext_vector_type(8)));

#define HID 128
#define A_STRIDE 132   // 128 max-K + pad to stagger LDS banks

// ---------------------------------------------------------------------------
// FP32 WMMA GEMM: Y[N x 128] = X[N x K] @ W[K x 128], K compile-time (9, 128).
// Block = 256 threads (8 waves); block computes a 16-row x 128-col strip.
// A tile (16 x K, zero-padded to KP=ceil4(K)) staged in LDS once; each wave
// owns one 16-col tile and runs the K-loop with V_WMMA_F32_16X16X4_F32.
// For K%4 != 0 (K=9) W is also staged zero-padded in LDS, so the MAC loop
// contains no guarded (EXEC-masked) loads for either K.
// N must be a multiple of 16 (100000 = 6250*16).
// ---------------------------------------------------------------------------
template <int K>
__global__ void __launch_bounds__(256)
gemm_wmma_f32(const float* __restrict__ X, const float* __restrict__ W,
              float* __restrict__ Y) {
    constexpr int KP = (K + 3) & ~3;
    __shared__ float ldsA[16 * A_STRIDE];
    __shared__ float ldsB[(K % 4 != 0) ? (KP * HID) : 4];

    const int tid   = threadIdx.x;
    const int mTile = blockIdx.x * 16;

    // ---- stage A tile (16 x KP), zero-pad cols K..KP-1 ----
    for (int i = tid; i < 16 * KP; i += 256) {
        const int r = i / KP, c = i - r * KP;
        ldsA[r * A_STRIDE + c] = (c < K) ? X[(size_t)(mTile + r) * K + c] : 0.0f;
    }
    if constexpr (K % 4 != 0) {
        // ---- stage W (KP x 128), zero-pad rows K..KP-1 ----
        for (int i = tid; i < KP * HID; i += 256) {
            const int k = i >> 7, c = i & (HID - 1);
            ldsB[i] = (k < K) ? W[(size_t)k * HID + c] : 0.0f;
        }
    }
    __syncthreads();

    const int wave = tid >> 5;          // 0..7 -> column tile
    const int lane = tid & 31;
    const int half = lane >> 4;         // 0: K=kb+0,1  1: K=kb+2,3
    const int l16  = lane & 15;
    const int koff = half * 2;
    const int nCol = wave * 16 + l16;

    v8f acc = {};
#pragma unroll ((K == 9) ? 3 : 8)
    for (int s = 0; s < (KP >> 2); ++s) {
        const int k0 = s * 4 + koff;
        // A fragment: row l16, cols k0..k0+1 (8B-aligned LDS read)
        v2f a = *(const v2f*)(&ldsA[l16 * A_STRIDE + k0]);
        v2f b;
        if constexpr (K % 4 != 0) {
            b.x = ldsB[k0 * HID + nCol];
            b.y = ldsB[(k0 + 1) * HID + nCol];
        } else {
            b.x = W[(size_t)k0 * HID + nCol];
            b.y = W[(size_t)(k0 + 1) * HID + nCol];
        }
        acc = __builtin_amdgcn_wmma_f32_16x16x4_f32(
            /*neg_a=*/false, a, /*neg_b=*/false, b,
            /*c_mod=*/(short)0, acc, /*reuse_a=*/false, /*reuse_b=*/false);
    }
    // D layout: VGPR r -> row (r + half*8), col l16 of this wave's tile
#pragma unroll
    for (int r = 0; r < 8; ++r) {
        Y[(size_t)(mTile + r + half * 8) * HID + nCol] = acc[r];
    }
}

// ---------------------------------------------------------------------------
// Degree: deg[dst] += 1 over edges, then dinv = rsqrt(deg + 1)  (self-loop)
// ---------------------------------------------------------------------------
__global__ void deg_accum(const int* __restrict__ dst, float* deg, int nE) {
    int e = blockIdx.x * blockDim.x + threadIdx.x;
    if (e < nE) atomicAdd(&deg[dst[e]], 1.0f);
}

__global__ void deg_finalize(float* deg, int n) {
    int i = blockIdx.x * blockDim.x + threadIdx.x;
    if (i < n) deg[i] = rsqrtf(deg[i] + 1.0f);
}

// ---------------------------------------------------------------------------
// Edge propagation: agg[dst] += h[src] * dinv[src]*dinv[dst]
// One thread per (edge, 4-feature chunk): float4 gather + 4 f32 atomics.
// ---------------------------------------------------------------------------
__global__ void __launch_bounds__(256)
edge_prop(const float* __restrict__ h, const int* __restrict__ src,
          const int* __restrict__ dst, const float* __restrict__ dinv,
          float* agg, int nE) {
    long long tid = (long long)blockIdx.x * blockDim.x + threadIdx.x;
    int e = (int)(tid >> 5);
    if (e >= nE) return;
    int f = ((int)tid & 31) << 2;                 // 0,4,...,124
    int s = src[e], d = dst[e];
    float c = dinv[s] * dinv[d];
    const float4 hv = *(const float4*)(h + (size_t)s * HID + f);
    float* o = agg + (size_t)d * HID + f;
    atomicAdd(o + 0, hv.x * c);
    atomicAdd(o + 1, hv.y * c);
    atomicAdd(o + 2, hv.z * c);
    atomicAdd(o + 3, hv.w * c);
}

// ---------------------------------------------------------------------------
// comb = agg + hW * dinv^2 + bias   (self-loop + bias), fused with the BN
// column reductions (per-block register accumulation, 1 atomic/col/block).
// blockDim.x == 128 (one thread per feature column).
// ---------------------------------------------------------------------------
__global__ void __launch_bounds__(128)
combine_reduce(const float* __restrict__ hw, const float* agg_in, float* comb,
               const float* __restrict__ dinv, const float* __restrict__ bias,
               float* colsum, float* colsq, int nNodes, int rowsPerBlock) {
    int f  = threadIdx.x;
    int r0 = blockIdx.x * rowsPerBlock;
    int r1 = r0 + rowsPerBlock; if (r1 > nNodes) r1 = nNodes;
    float b = bias[f];
    float s = 0.0f, sq = 0.0f;
    for (int r = r0; r < r1; ++r) {
        float di = dinv[r];
        size_t i = (size_t)r * HID + f;
        float v = agg_in[i] + hw[i] * di * di + b;
        comb[i] = v;
        s  += v;
        sq += v * v;
    }
    atomicAdd(&colsum[f], s);
    atomicAdd(&colsq[f], sq);
}

// ---------------------------------------------------------------------------
// BatchNorm (biased var) + ReLU, elementwise over N x 128.
// ---------------------------------------------------------------------------
__global__ void __launch_bounds__(256)
bn_relu(const float* __restrict__ comb, float* __restrict__ out,
        const float* __restrict__ colsum, const float* __restrict__ colsq,
        const float* __restrict__ g, const float* __restrict__ beta, int nNodes) {
    size_t i = (size_t)blockIdx.x * blockDim.x + threadIdx.x;
    if (i >= (size_t)nNodes * HID) return;
    int f = (int)(i & (HID - 1));
    float invN = 1.0f / (float)nNodes;
    float mean = colsum[f] * invN;
    float var  = colsq[f] * invN - mean * mean;
    float xn = (comb[i] - mean) * rsqrtf(var + 1e-5f);
    float v  = xn * g[f] + beta[f];
    out[i] = v > 0.0f ? v : 0.0f;
}

// ---------------------------------------------------------------------------
// Pooling: mean-sum + count + max (h >= 0 post-ReLU, so uint-bit max valid;
// zero-init == 0.0f matches reference's isfinite fixup for empty segments).
// ---------------------------------------------------------------------------
__global__ void __launch_bounds__(256)
pool_accum(const float* __restrict__ h, const int* __restrict__ batch,
           float* psum, unsigned int* pmax, float* cnt, int nNodes) {
    size_t tid = (size_t)blockIdx.x * blockDim.x + threadIdx.x;
    int node = (int)(tid >> 7);
    if (node >= nNodes) return;
    int f = (int)(tid & (HID - 1));
    int g = batch[node];
    float v = h[tid];
    atomicAdd(&psum[(size_t)g * HID + f], v);
    atomicMax(&pmax[(size_t)g * HID + f], __float_as_uint(v));   // v >= 0
    if (f == 0) atomicAdd(&cnt[g], 1.0f);
}

__global__ void __launch_bounds__(256)
pool_final(const float* __restrict__ psum, const unsigned int* __restrict__ pmax,
           const float* __restrict__ cnt, float* __restrict__ out, int nGraphs) {
    int tid = blockIdx.x * blockDim.x + threadIdx.x;
    if (tid >= nGraphs * HID) return;
    int g = tid >> 7, f = tid & (HID - 1);
    float c = cnt[g]; if (c < 1.0f) c = 1.0f;
    out[(size_t)g * (2 * HID) + f]       = psum[tid] / c;
    out[(size_t)g * (2 * HID) + HID + f] = __uint_as_float(pmax[tid]);
}

// ---------------------------------------------------------------------------
// Launcher.  Workspace layout (floats):
//   T [N*128] ping, A [N*128] pong, dinv [N], colsum/colsq [128 each],
//   psum [G*128], pmax [G*128] (uint), cnt [G].   ~103 MB total.
// ---------------------------------------------------------------------------
extern "C" void kernel_launch(void* const* d_in, const int* in_sizes, int n_in,
                              void* d_out, int out_size, void* d_ws, size_t ws_size,
                              hipStream_t stream) {
    const float* x     = (const float*)d_in[0];
    const int*   ei    = (const int*)d_in[1];
    const int*   batch = (const int*)d_in[2];
    const float* W1 = (const float*)d_in[3],  *b1 = (const float*)d_in[4];
    const float* g1 = (const float*)d_in[5],  *be1 = (const float*)d_in[6];
    const float* W2 = (const float*)d_in[7],  *b2 = (const float*)d_in[8];
    const float* g2 = (const float*)d_in[9],  *be2 = (const float*)d_in[10];
    const float* W3 = (const float*)d_in[11], *b3 = (const float*)d_in[12];
    const float* g3 = (const float*)d_in[13], *be3 = (const float*)d_in[14];

    const int N = in_sizes[0] / 9;          // 100000
    const int E = in_sizes[1] / 2;          // 1600000
    const int G = out_size / (2 * HID);     // 256

    const int* src = ei;
    const int* dst = ei + E;

    const size_t NH = (size_t)N * HID;
    float* T      = (float*)d_ws;
    float* A      = T + NH;
    float* dinv   = A + NH;
    float* colsum = dinv + N;
    float* colsq  = colsum + HID;
    float* psum   = colsq + HID;
    unsigned int* pmax = (unsigned int*)(psum + (size_t)G * HID);
    float* cnt    = (float*)(pmax + (size_t)G * HID);

    // ---- degree / symmetric normalization coefficients ----
    hipMemsetAsync(dinv, 0, (size_t)N * sizeof(float), stream);
    deg_accum<<<(E + 255) / 256, 256, 0, stream>>>(dst, dinv, E);
    deg_finalize<<<(N + 255) / 256, 256, 0, stream>>>(dinv, N);

    const long long edgeThreads = (long long)E * 32;
    const int edgeBlocks = (int)((edgeThreads + 255) / 256);
    const int gemmBlocks = N / 16;

    // One layer (stream-ordered, so xin's buffer is reusable after the gemm):
    //   gemm   : xin -> Gb
    //   prop   : Gb  -> Sb (agg, zeroed)
    //   combine: (Gb, Sb) -> Sb
    //   bn_relu: Sb -> Gb            => layer output lives in Gb
    auto layer = [&](const float* xin, int K, float* Gb, float* Sb,
                     const float* W, const float* b, const float* g, const float* beta) {
        if (K == 9)
            gemm_wmma_f32<9><<<gemmBlocks, 256, 0, stream>>>(xin, W, Gb);
        else
            gemm_wmma_f32<128><<<gemmBlocks, 256, 0, stream>>>(xin, W, Gb);
        hipMemsetAsync(Sb, 0, NH * sizeof(float), stream);
        edge_prop<<<edgeBlocks, 256, 0, stream>>>(Gb, src, dst, dinv, Sb, E);
        hipMemsetAsync(colsum, 0, 2 * HID * sizeof(float), stream);
        const int rpb = 64;
        combine_reduce<<<(N + rpb - 1) / rpb, 128, 0, stream>>>(
            Gb, Sb, Sb, dinv, b, colsum, colsq, N, rpb);
        bn_relu<<<(int)((NH + 255) / 256), 256, 0, stream>>>(
            Sb, Gb, colsum, colsq, g, beta, N);
    };

    layer(x, 9,   T, A, W1, b1, g1, be1);    // out in T
    layer(T, 128, A, T, W2, b2, g2, be2);    // out in A
    layer(A, 128, T, A, W3, b3, g3, be3);    // out in T

    // ---- global mean / max pooling ----
    hipMemsetAsync(psum, 0, (size_t)G * HID * sizeof(float), stream);
    hipMemsetAsync(pmax, 0, (size_t)G * HID * sizeof(unsigned int), stream);
    hipMemsetAsync(cnt,  0, (size_t)G * sizeof(float), stream);
    pool_accum<<<(int)((NH + 255) / 256), 256, 0, stream>>>(T, batch, psum, pmax, cnt, N);
    pool_final<<<(G * HID + 255) / 256, 256, 0, stream>>>(psum, pmax, cnt, (float*)d_out, G);
}